// KPConv_59528246722702
// MI455X (gfx1250) — compile-verified
//
#include <hip/hip_runtime.h>

typedef float v2f __attribute__((ext_vector_type(2)));
typedef float v8f __attribute__((ext_vector_type(8)));

namespace {
constexpr int B  = 4;
constexpr int N  = 8192;
constexpr int M  = 8192;
constexpr int NB = 32;
constexpr int KP = 15;              // kernel points (padded to 16 in WMMA M)
constexpr int C  = 64;              // input channels
constexpr int D  = 64;              // output channels
constexpr int KK = KP * C;          // 960 : GEMM2 contraction length
constexpr int WSTR = KK + 2;        // 962-dword LDS row stride (8B aligned, bank spread)
constexpr float SIGMA_INV = 0.5f;   // 1/SIGMA
}

__global__ __launch_bounds__(128)
void kpconv_wmma_f32(const float* __restrict__ q_points,   // (B,N,3)
                     const float* __restrict__ s_points,   // (B,M,3)
                     const float* __restrict__ s_feats,    // (B,M,C)
                     const int*   __restrict__ nbrs,       // (B,N,NB)
                     const float* __restrict__ kpts,       // (KP,3)
                     const float* __restrict__ wts,        // (KP,C,D) == (KK,D)
                     float* __restrict__ out)              // (B,N,D)
{
  __shared__ float  lds_w[16 * WSTR];    // weighted: 16 points x 960 (+pad)
  __shared__ float4 lds_xyz[4][NB];      // per-wave neighbor offsets
  __shared__ int    lds_idx[4][NB];      // per-wave clamped neighbor indices
  __shared__ float  lds_nnum[16];        // per-point normalizer

  const int tid  = threadIdx.x;
  const int wv   = tid >> 5;             // wave id 0..3
  const int lane = tid & 31;
  const int h    = lane >> 4;            // half-wave (selects K sub-columns)
  const int nlo  = lane & 15;

  const int tile = blockIdx.x;           // B*N/16 tiles
  const int b    = tile / (N / 16);
  const int base = (tile - b * (N / 16)) * 16;

  const float* __restrict__ sfb = s_feats + (size_t)b * M * C;

  // this lane's kernel-point (WMMA A-matrix row); row 15 is zero padding
  const int   mcl = (nlo < KP) ? nlo : (KP - 1);
  const float kx = kpts[mcl * 3 + 0];
  const float ky = kpts[mcl * 3 + 1];
  const float kz = kpts[mcl * 3 + 2];
  const float mrow_scale = (nlo < KP) ? 1.0f : 0.0f;

  const v8f vzero = {0.f, 0.f, 0.f, 0.f, 0.f, 0.f, 0.f, 0.f};

  // ---------- phase 1: per point, weighted(16x64) = w(16x32) @ feats(32x64)
  for (int p2 = 0; p2 < 4; ++p2) {
    const int p = wv * 4 + p2;
    const int n = base + p;
    __syncthreads();                     // protect staging reuse (all waves aligned)

    // lane <-> neighbor slot
    const int  idx   = nbrs[((size_t)b * N + n) * NB + lane];
    const bool valid = idx < M;          // idx==M is the INF/zero sentinel
    const int  cidx  = valid ? idx : (M - 1);

    const float qx = q_points[((size_t)b * N + n) * 3 + 0];
    const float qy = q_points[((size_t)b * N + n) * 3 + 1];
    const float qz = q_points[((size_t)b * N + n) * 3 + 2];
    const float* sp = s_points + ((size_t)b * M + cidx) * 3;
    const float px = valid ? sp[0] : 1.0e6f;   // far away -> w clamps to 0
    const float py = valid ? sp[1] : 1.0e6f;
    const float pz = valid ? sp[2] : 1.0e6f;
    lds_xyz[wv][lane] = make_float4(px - qx, py - qy, pz - qz, 0.0f);
    lds_idx[wv][lane] = cidx;

    // neighbor feature row-sum (normalizer); zero-pad rows never count
    const float4* fr = (const float4*)(sfb + (size_t)cidx * C);
    float rs = 0.0f;
    #pragma unroll
    for (int i = 0; i < C / 4; ++i) {
      float4 v = fr[i];
      rs += (v.x + v.y) + (v.z + v.w);
    }
    const unsigned long long bal = __ballot(valid && (rs > 0.0f));
    if (lane == 0) {
      const int cnt = (int)__popcll(bal);
      lds_nnum[p] = (float)(cnt > 0 ? cnt : 1);
    }
    __syncthreads();

    // the 16 w values this lane feeds into A-fragments:
    // A(16x4) layout: lane m=nlo, a = { col 4s+2h, col 4s+2h+1 }
    float wloc[16];
    #pragma unroll
    for (int s = 0; s < 8; ++s) {
      #pragma unroll
      for (int j = 0; j < 2; ++j) {
        const float4 dp = lds_xyz[wv][4 * s + 2 * h + j];
        const float dx = dp.x - kx, dy = dp.y - ky, dz = dp.z - kz;
        const float sq = dx * dx + dy * dy + dz * dz;
        wloc[2 * s + j] =
            fmaxf(0.0f, 1.0f - sqrtf(sq) * SIGMA_INV) * mrow_scale;
      }
    }

    v8f acc[4];
    #pragma unroll
    for (int t = 0; t < 4; ++t) acc[t] = vzero;

    #pragma unroll
    for (int s = 0; s < 8; ++s) {
      const int r0 = 4 * s + 2 * h;               // B-matrix rows for this half
      const int i0 = lds_idx[wv][r0];
      const int i1 = lds_idx[wv][r0 + 1];
      const float* f0 = sfb + (size_t)i0 * C + nlo;
      const float* f1 = sfb + (size_t)i1 * C + nlo;
      v2f a;
      a.x = wloc[2 * s + 0];
      a.y = wloc[2 * s + 1];
      #pragma unroll
      for (int t = 0; t < 4; ++t) {               // C tiles of 16
        v2f bf;
        bf.x = f0[16 * t];
        bf.y = f1[16 * t];
        acc[t] = __builtin_amdgcn_wmma_f32_16x16x4_f32(
            false, a, false, bf, (short)0, acc[t], false, false);
      }
    }

    // D layout: lane holds D[m = v+8h][n = 16t+nlo]; scatter rows m<15 to LDS
    #pragma unroll
    for (int t = 0; t < 4; ++t) {
      #pragma unroll
      for (int v = 0; v < 8; ++v) {
        const int mr = v + 8 * h;
        if (mr < KP)
          lds_w[p * WSTR + mr * C + 16 * t + nlo] = acc[t][v];
      }
    }
  }
  __syncthreads();

  // ---------- phase 2: out(16xD) = weighted(16x960) @ wts(960xD)
  // wave wv owns D columns [16*wv, 16*wv+16); 4 round-robin accumulators
  const float* aw = &lds_w[nlo * WSTR + 2 * h];                 // A rows: point=nlo
  const float* wp = wts + (size_t)(2 * h) * D + 16 * wv + nlo;  // B col base

  v8f oacc[4];
  #pragma unroll
  for (int u = 0; u < 4; ++u) oacc[u] = vzero;

  #pragma unroll 2
  for (int s = 0; s < KK / 16; ++s) {       // 60 x 4 = 240 k-steps of 4
    #pragma unroll
    for (int u = 0; u < 4; ++u) {
      const int s2 = 4 * s + u;
      const v2f a = *(const v2f*)(aw + 4 * s2);     // 8B-aligned ds_load_b64
      v2f bf;
      bf.x = wp[(size_t)(4 * s2) * D];
      bf.y = wp[(size_t)(4 * s2 + 1) * D];
      oacc[u] = __builtin_amdgcn_wmma_f32_16x16x4_f32(
          false, a, false, bf, (short)0, oacc[u], false, false);
    }
  }

  // normalize and store: lane holds out[point = v+8h][d = 16*wv + nlo]
  float rn[8];
  #pragma unroll
  for (int v = 0; v < 8; ++v) rn[v] = 1.0f / lds_nnum[v + 8 * h];

  float* ob = out + ((size_t)b * N + base) * D + 16 * wv + nlo;
  #pragma unroll
  for (int v = 0; v < 8; ++v) {
    const int prow = v + 8 * h;
    const float val =
        ((oacc[0][v] + oacc[1][v]) + (oacc[2][v] + oacc[3][v])) * rn[v];
    ob[(size_t)prow * D] = val;
  }
}

extern "C" void kernel_launch(void* const* d_in, const int* in_sizes, int n_in,
                              void* d_out, int out_size, void* d_ws, size_t ws_size,
                              hipStream_t stream) {
  (void)in_sizes; (void)n_in; (void)out_size; (void)d_ws; (void)ws_size;
  const float* q_points = (const float*)d_in[0];
  const float* s_points = (const float*)d_in[1];
  // d_in[2] (q_feats) is unused by the reference
  const float* s_feats  = (const float*)d_in[3];
  const int*   nbrs     = (const int*)d_in[4];
  const float* kpts     = (const float*)d_in[5];
  const float* wts      = (const float*)d_in[6];
  float* out = (float*)d_out;

  dim3 grid((B * N) / 16);   // 2048 tiles of 16 query points
  dim3 block(128);           // 4 waves: 4 points each in phase 1, 1 D-tile each in phase 2
  kpconv_wmma_f32<<<grid, block, 0, stream>>>(q_points, s_points, s_feats,
                                              nbrs, kpts, wts, out);
}